// GNN_10831907520707
// MI455X (gfx1250) — compile-verified
//
#include <hip/hip_runtime.h>
#include <hip/hip_bf16.h>

typedef __attribute__((ext_vector_type(2))) float v2f;
typedef __attribute__((ext_vector_type(8))) float v8f;

#define H 64  // embedding width

// ---------------------------------------------------------------- utilities
__global__ void gcn_zero_f32(float* __restrict__ p, int n) {
  int i = blockIdx.x * blockDim.x + threadIdx.x;
  if (i < n) p[i] = 0.0f;
}

__global__ void gcn_degree(const int* __restrict__ dst, float* __restrict__ deg, int nE) {
  int e = blockIdx.x * blockDim.x + threadIdx.x;
  if (e < nE) atomicAdd(&deg[dst[e]], 1.0f);
}

__global__ void gcn_dinv(float* __restrict__ d, int nN) {
  int i = blockIdx.x * blockDim.x + threadIdx.x;
  if (i < nN) d[i] = rsqrtf(d[i] + 1.0f);  // +1 for self-loop
}

__global__ void gcn_norm(const int* __restrict__ src, const int* __restrict__ dst,
                         const float* __restrict__ dinv, float* __restrict__ norm, int nE) {
  int e = blockIdx.x * blockDim.x + threadIdx.x;
  if (e < nE) norm[e] = dinv[src[e]] * dinv[dst[e]];
}

// ---------------------------------------------------------------- GEMM: Y = X @ W
// One wave computes a 16-row x 64-col tile using V_WMMA_F32_16X16X4_F32.
// A (16x4 f32): lane L holds row M=L&15; VGPR j holds K = 2*(L>>4) + j  -> aligned float2 load.
// B (4x16 f32): lane L holds col N=L&15; VGPR j holds K = 2*(L>>4) + j.
// C/D (16x16):  VGPR r, lane L -> M = r + 8*(L>>4), N = L&15.
__global__ void __launch_bounds__(256)
gcn_gemm(const float* __restrict__ X, const float* __restrict__ W,
         float* __restrict__ Y, int n_tiles) {
  const int wave = threadIdx.x >> 5;
  const int lane = threadIdx.x & 31;
  const int tile = blockIdx.x * 8 + wave;
  if (tile >= n_tiles) return;  // wave-uniform: EXEC stays all-ones for WMMA

  const int row0  = tile << 4;
  const int m     = lane & 15;
  const int khalf = lane >> 4;

  const float* __restrict__ xrow = X + (size_t)(row0 + m) * H + 2 * khalf;

  v8f acc0 = {}, acc1 = {}, acc2 = {}, acc3 = {};

#pragma unroll
  for (int kt = 0; kt < 16; ++kt) {
    const float2 av = *(const float2*)(xrow + 4 * kt);
    v2f A; A.x = av.x; A.y = av.y;

    const int kb = 4 * kt + 2 * khalf;
    const float* __restrict__ wp = W + (size_t)kb * H + m;
    v2f B0, B1, B2, B3;
    B0.x = wp[0];   B0.y = wp[64];
    B1.x = wp[16];  B1.y = wp[80];
    B2.x = wp[32];  B2.y = wp[96];
    B3.x = wp[48];  B3.y = wp[112];

    acc0 = __builtin_amdgcn_wmma_f32_16x16x4_f32(false, A, false, B0, (short)0, acc0, false, false);
    acc1 = __builtin_amdgcn_wmma_f32_16x16x4_f32(false, A, false, B1, (short)0, acc1, false, false);
    acc2 = __builtin_amdgcn_wmma_f32_16x16x4_f32(false, A, false, B2, (short)0, acc2, false, false);
    acc3 = __builtin_amdgcn_wmma_f32_16x16x4_f32(false, A, false, B3, (short)0, acc3, false, false);
  }

  const int nb    = lane & 15;
  const int mhalf = lane >> 4;
#pragma unroll
  for (int r = 0; r < 8; ++r) {
    const size_t o = (size_t)(row0 + r + 8 * mhalf) * H + nb;
    Y[o +  0] = acc0[r];
    Y[o + 16] = acc1[r];
    Y[o + 32] = acc2[r];
    Y[o + 48] = acc3[r];
  }
}

// ---------------------------------------------------------------- out = xw * dinv^2 + b
// 16 threads per node, float4 per thread.
__global__ void gcn_self_bias(const float* __restrict__ xw, const float* __restrict__ dinv,
                              const float* __restrict__ b, float* __restrict__ out, int nN) {
  int t = blockIdx.x * blockDim.x + threadIdx.x;
  int node = t >> 4;
  if (node >= nN) return;
  int q = (t & 15) * 4;
  float di = dinv[node];
  float sn = di * di;
  float4 v  = *(const float4*)(xw + (size_t)node * H + q);
  float4 bb = *(const float4*)(b + q);
  float4 r;
  r.x = v.x * sn + bb.x;
  r.y = v.y * sn + bb.y;
  r.z = v.z * sn + bb.z;
  r.w = v.w * sn + bb.w;
  *(float4*)(out + (size_t)node * H + q) = r;
}

// ---------------------------------------------------------------- edge scatter-add
// 16 threads per edge; gather float4 from xw[src] (L2-resident), scale by norm,
// atomic-add into out[dst].
__global__ void gcn_scatter(const float* __restrict__ xw, const int* __restrict__ src,
                            const int* __restrict__ dst, const float* __restrict__ norm,
                            float* __restrict__ out, int nE) {
  int t = blockIdx.x * blockDim.x + threadIdx.x;
  int e = t >> 4;
  if (e >= nE) return;
  int q = (t & 15) * 4;
  int s = src[e];
  int d = dst[e];
  float w = norm[e];
  float4 v = *(const float4*)(xw + (size_t)s * H + q);
  float* o = out + (size_t)d * H + q;
  atomicAdd(o + 0, v.x * w);
  atomicAdd(o + 1, v.y * w);
  atomicAdd(o + 2, v.z * w);
  atomicAdd(o + 3, v.w * w);
}

// ---------------------------------------------------------------- launcher
extern "C" void kernel_launch(void* const* d_in, const int* in_sizes, int n_in,
                              void* d_out, int out_size, void* d_ws, size_t ws_size,
                              hipStream_t stream) {
  const float* emb = (const float*)d_in[0];
  const int*   ei  = (const int*)d_in[1];
  const float* W1  = (const float*)d_in[2];
  const float* b1  = (const float*)d_in[3];
  const float* W2  = (const float*)d_in[4];
  const float* b2  = (const float*)d_in[5];
  const float* W3  = (const float*)d_in[6];
  const float* b3  = (const float*)d_in[7];

  const int N = in_sizes[0] / H;       // 100000
  const int E = in_sizes[1] / 2;       // 1600000
  const int* src = ei;
  const int* dst = ei + E;

  // workspace layout (256B aligned)
  auto align256 = [](size_t x) { return (x + 255) & ~(size_t)255; };
  char* ws = (char*)d_ws;
  size_t off = 0;
  float* dinv = (float*)(ws + off); off = align256(off + (size_t)N * sizeof(float));
  float* norm = (float*)(ws + off); off = align256(off + (size_t)E * sizeof(float));
  float* xw   = (float*)(ws + off); off = align256(off + (size_t)N * H * sizeof(float));
  float* xbuf = (float*)(ws + off); off = align256(off + (size_t)N * H * sizeof(float));
  (void)ws_size;

  const int TB = 256;
  const int gN   = (N + TB - 1) / TB;
  const int gE   = (E + TB - 1) / TB;
  const int n_tiles = N / 16;                       // N divisible by 16
  const int gGemm = (n_tiles + 7) / 8;              // 8 waves per block
  const int gNode16 = ((size_t)N * 16 + TB - 1) / TB;
  const int gEdge16 = ((size_t)E * 16 + TB - 1) / TB;

  float* out = (float*)d_out;

  // ---- degree / norm precompute (shared by all 3 layers)
  gcn_zero_f32<<<gN, TB, 0, stream>>>(dinv, N);
  gcn_degree  <<<gE, TB, 0, stream>>>(dst, dinv, E);
  gcn_dinv    <<<gN, TB, 0, stream>>>(dinv, N);
  gcn_norm    <<<gE, TB, 0, stream>>>(src, dst, dinv, norm, E);

  // ---- layer 1: emb -> xbuf
  gcn_gemm     <<<gGemm,   TB, 0, stream>>>(emb, W1, xw, n_tiles);
  gcn_self_bias<<<gNode16, TB, 0, stream>>>(xw, dinv, b1, xbuf, N);
  gcn_scatter  <<<gEdge16, TB, 0, stream>>>(xw, src, dst, norm, xbuf, E);

  // ---- layer 2: xbuf -> xbuf (gemm consumes xbuf before it is overwritten)
  gcn_gemm     <<<gGemm,   TB, 0, stream>>>(xbuf, W2, xw, n_tiles);
  gcn_self_bias<<<gNode16, TB, 0, stream>>>(xw, dinv, b2, xbuf, N);
  gcn_scatter  <<<gEdge16, TB, 0, stream>>>(xw, src, dst, norm, xbuf, E);

  // ---- layer 3: xbuf -> d_out
  gcn_gemm     <<<gGemm,   TB, 0, stream>>>(xbuf, W3, xw, n_tiles);
  gcn_self_bias<<<gNode16, TB, 0, stream>>>(xw, dinv, b3, out, N);
  gcn_scatter  <<<gEdge16, TB, 0, stream>>>(xw, src, dst, norm, out, E);
}